// SpectralConv1d_41540923687463
// MI455X (gfx1250) — compile-verified
//
#include <hip/hip_runtime.h>

// ---------------------------------------------------------------------------
// SpectralConv1d via truncated DHT, all heavy lifting on v_wmma_f32_16x16x4_f32
//   B=32, Ci=Co=128, N=8192, modes=128, M=N/2+1=4097
//   a  = X(4096x8192) @ C1(8192x128)          C1[n,k] = cas(2*pi*n*k/N)
//   Z  = per-mode mix with even/odd weights (flip folded into weights)
//   out= Z(4096x128)  @ C2(128x4097)          C2[k,n] = cas(2*pi*k*n/M)/M
// ---------------------------------------------------------------------------

typedef float v2f __attribute__((ext_vector_type(2)));
typedef float v8f __attribute__((ext_vector_type(8)));

#define B_    32
#define CI    128
#define CO    128
#define NN    8192
#define MODES 128
#define MM    4097      // N/2 + 1
#define NPAD  4112      // MM rounded up to 16

__device__ __forceinline__ v8f wmma_f32(v2f a, v2f b, v8f c) {
    // (neg_a, A, neg_b, B, c_mod, C, reuse_a, reuse_b)
    return __builtin_amdgcn_wmma_f32_16x16x4_f32(false, a, false, b,
                                                 (short)0, c, false, false);
}

// --------------------------- basis builders --------------------------------

__global__ __launch_bounds__(256) void build_c1(float* __restrict__ C1) {
    const int idx = blockIdx.x * 256 + threadIdx.x;     // < 8192*128
    const int n = idx >> 7;
    const int k = idx & 127;
    const int ph = (n * k) & (NN - 1);                  // exact mod 8192
    const float ang = (float)ph * (6.28318530717958647692f / (float)NN);
    float s, c;
    __sincosf(ang, &s, &c);
    C1[idx] = c + s;                                    // cas()
}

__global__ __launch_bounds__(256) void build_c2(float* __restrict__ C2) {
    const int idx = blockIdx.x * 256 + threadIdx.x;     // < 128*4112 (exact)
    const int k = idx / NPAD;
    const int n = idx % NPAD;
    float v = 0.0f;
    if (n < MM) {
        const int ph = (k * n) % MM;                    // exact in int
        const float ang = (float)ph * (6.28318530717958647692f / (float)MM);
        float s, c;
        __sincosf(ang, &s, &c);
        v = (c + s) * (1.0f / (float)MM);
    }
    C2[idx] = v;
}

// ------------------- GEMM1: a = X @ C1  (4096x8192x128) --------------------
// grid.x = 256 row strips of 16; 8 waves = 8 column tiles of 16 (128 cols)

__global__ __launch_bounds__(256) void dht_fwd_gemm(const float* __restrict__ X,
                                                    const float* __restrict__ C1,
                                                    float* __restrict__ A)
{
    constexpr int KC = 64;
    __shared__ float lA[16 * 68];        // 16 x 64, stride 68 (bank-safe, 16B ok)
    __shared__ float lB[KC * 128];       // 64 x 128 chunk of C1 (32 KB)

    const int rowBase = blockIdx.x * 16;
    const int t    = threadIdx.x;
    const int lane = t & 31;
    const int wave = t >> 5;
    const int colBase = wave * 16;
    const int lrow = lane & 15;
    const int kh   = (lane >> 4) << 1;   // 0 or 2 (K half per ISA layout)

    v8f acc = {};
    for (int k0 = 0; k0 < NN; k0 += KC) {
        {   // stage X strip chunk: 16 rows x 64, float4 per thread
            const int r  = t >> 4;           // 0..15
            const int c4 = (t & 15) << 2;    // 0..60
            const float4 v = *reinterpret_cast<const float4*>(
                &X[(size_t)(rowBase + r) * NN + k0 + c4]);
            *reinterpret_cast<float4*>(&lA[r * 68 + c4]) = v;
        }
        #pragma unroll
        for (int j = 0; j < 8; ++j) {    // stage C1 chunk: 64 x 128
            const int f4 = t + j * 256;      // float4 index 0..2047
            const int r  = f4 >> 5;          // 0..63
            const int c  = (f4 & 31) << 2;   // 0..124
            const float4 v = *reinterpret_cast<const float4*>(
                &C1[(size_t)(k0 + r) * MODES + c]);
            *reinterpret_cast<float4*>(&lB[r * 128 + c]) = v;
        }
        __syncthreads();
        #pragma unroll
        for (int kk = 0; kk < KC; kk += 4) {
            v2f a, b;
            a.x = lA[lrow * 68 + kk + kh];
            a.y = lA[lrow * 68 + kk + kh + 1];
            const float* bp = &lB[(kk + kh) * 128 + colBase + lrow];
            b.x = bp[0];
            b.y = bp[128];
            acc = wmma_f32(a, b, acc);
        }
        __syncthreads();
    }
    #pragma unroll
    for (int r = 0; r < 8; ++r) {
        const int row = rowBase + r + ((lane >> 4) << 3);
        A[(size_t)row * MODES + colBase + lrow] = acc[r];
    }
}

// ------------------- mode mixing (flip folded into weights) ----------------
// Z[b,o,k] = 0.5 * sum_i [ a[b,i,k]*(W[i,o,k]+W[i,o,kf])
//                        + a[b,i,kf]*(W[i,o,k]-W[i,o,kf]) ],  kf=(-k) mod 128
// one block per k; threads = (o, b-half); a slices staged in LDS

__global__ __launch_bounds__(256) void mix_modes(const float* __restrict__ A,
                                                 const float* __restrict__ W,
                                                 float* __restrict__ Z)
{
    const int k  = blockIdx.x;                 // 0..127
    const int kf = (MODES - k) & (MODES - 1);
    __shared__ float sa[B_ * CI];              // a[b][i] at mode k   (16 KB)
    __shared__ float sf[B_ * CI];              // a[b][i] at mode kf  (16 KB)
    const int t = threadIdx.x;
    for (int idx = t; idx < B_ * CI; idx += 256) {
        sa[idx] = A[(size_t)idx * MODES + k];
        sf[idx] = A[(size_t)idx * MODES + kf];
    }
    __syncthreads();

    const int o  = t & 127;
    const int bh = t >> 7;                     // 0 or 1 -> 16 b's each
    float acc[16];
    #pragma unroll
    for (int j = 0; j < 16; ++j) acc[j] = 0.0f;

    for (int i = 0; i < CI; ++i) {
        const size_t wbase = ((size_t)i * CO + o) * MODES;
        const float w1 = W[wbase + k];
        const float w2 = W[wbase + kf];
        const float we = 0.5f * (w1 + w2);
        const float wo = 0.5f * (w1 - w2);
        #pragma unroll
        for (int j = 0; j < 16; ++j) {
            const int b = bh * 16 + j;
            acc[j] += sa[b * CI + i] * we + sf[b * CI + i] * wo;
        }
    }
    #pragma unroll
    for (int j = 0; j < 16; ++j) {
        const int b = bh * 16 + j;
        Z[((size_t)b * CO + o) * MODES + k] = acc[j];
    }
}

// ------------------- GEMM2: out = Z @ C2  (4096x128x4097) ------------------
// grid = (256 row strips, 33 col groups of 128); ragged edge guarded on store

__global__ __launch_bounds__(256) void idht_gemm(const float* __restrict__ Z,
                                                 const float* __restrict__ C2,
                                                 float* __restrict__ out)
{
    __shared__ float lZ[16 * 132];       // 16 x 128, stride 132 (bank-safe)
    const int rowBase = blockIdx.x * 16;
    const int t    = threadIdx.x;
    const int lane = t & 31;
    const int wave = t >> 5;
    const int colBase = blockIdx.y * 128 + wave * 16;
    const int lrow = lane & 15;
    const int kh   = (lane >> 4) << 1;

    {   // stage Z strip: 16 x 128, two float4 per thread
        const int r  = t >> 4;
        const int c8 = (t & 15) << 3;
        const float4* src = reinterpret_cast<const float4*>(
            &Z[(size_t)(rowBase + r) * MODES + c8]);
        *reinterpret_cast<float4*>(&lZ[r * 132 + c8])     = src[0];
        *reinterpret_cast<float4*>(&lZ[r * 132 + c8 + 4]) = src[1];
    }
    __syncthreads();
    if (colBase >= MM) return;           // wave-uniform: EXEC stays all-ones

    v8f acc = {};
    #pragma unroll
    for (int kk = 0; kk < MODES; kk += 4) {
        v2f a, b;
        a.x = lZ[lrow * 132 + kk + kh];
        a.y = lZ[lrow * 132 + kk + kh + 1];
        const float* bp = &C2[(size_t)(kk + kh) * NPAD + colBase + lrow];
        b.x = bp[0];
        b.y = bp[NPAD];
        acc = wmma_f32(a, b, acc);
    }
    const int col = colBase + lrow;
    if (col < MM) {
        #pragma unroll
        for (int r = 0; r < 8; ++r) {
            const int row = rowBase + r + ((lane >> 4) << 3);
            out[(size_t)row * MM + col] = acc[r];
        }
    }
}

// ---------------------------------------------------------------------------

extern "C" void kernel_launch(void* const* d_in, const int* in_sizes, int n_in,
                              void* d_out, int out_size, void* d_ws, size_t ws_size,
                              hipStream_t stream) {
    (void)in_sizes; (void)n_in; (void)out_size; (void)ws_size;
    const float* x = (const float*)d_in[0];      // (32,128,8192) f32
    const float* w = (const float*)d_in[1];      // (128,128,128) f32
    float* out = (float*)d_out;                  // (32,128,4097) f32

    float* C1 = (float*)d_ws;                    // 8192*128   = 1,048,576 f
    float* A  = C1 + (size_t)NN * MODES;         // 4096*128   =   524,288 f
    float* Zb = A  + (size_t)(B_ * CI) * MODES;  // 4096*128   =   524,288 f
    float* C2 = Zb + (size_t)(B_ * CO) * MODES;  // 128*4112   =   526,336 f
                                                 // total ~10.5 MB workspace

    build_c1<<<(NN * MODES) / 256, 256, 0, stream>>>(C1);
    build_c2<<<(MODES * NPAD) / 256, 256, 0, stream>>>(C2);
    dht_fwd_gemm<<<(B_ * CI) / 16, 256, 0, stream>>>(x, C1, A);
    mix_modes<<<MODES, 256, 0, stream>>>(A, w, Zb);
    idht_gemm<<<dim3((B_ * CO) / 16, (NPAD + 127) / 128), 256, 0, stream>>>(Zb, C2, out);
}